// ISTFTLayer_62302795596300
// MI455X (gfx1250) — compile-verified
//
#include <hip/hip_runtime.h>
#include <math.h>

// ---------------------------------------------------------------------------
// ISTFT as a single fp32 WMMA GEMM on gfx1250 (MI455X).
//   frames_windowed = [Re | Im] @ Basis          (M=32768, K=1032, N=1024)
// Basis folds: irfft cos/sin terms, 2/N & 1/N coefficients, DC + Nyquist
// handling, and the COLA-normalized Hann window (hann/1.5). Overlap-add is
// done with global fp32 atomics into the trimmed output.
// A-tiles are staged into LDS by the Tensor Data Mover (TDM) with hardware
// LDS padding (row stride 132 floats), double-buffered against the WMMA loop.
// ---------------------------------------------------------------------------

typedef float v2f  __attribute__((ext_vector_type(2)));
typedef float v8f  __attribute__((ext_vector_type(8)));
typedef unsigned int u32x4 __attribute__((ext_vector_type(4)));
typedef int   i32x8 __attribute__((ext_vector_type(8)));
typedef int   i32x4 __attribute__((ext_vector_type(4)));

#define B_N      64
#define FRAMES   512
#define BINS     513
#define NFFT     1024
#define STRIDE_  256
#define KCOLS    1032          // 516 (real block) + 516 (imag block)
#define KCHUNKS  258           // KCOLS / 4  (K-steps of the 16x16x4 WMMA)
#define OUT_T    130816        // 256*511 + 1024 - 1024  (trimmed length)
#define NTILES   64            // 1024 / 16 n-tiles
#define TWO_PI   6.28318530717958647692f

#if __has_builtin(__builtin_amdgcn_tensor_load_to_lds) && __has_builtin(__builtin_amdgcn_s_wait_tensorcnt)
#define HAVE_TDM 1
#else
#define HAVE_TDM 0
#endif

// Basis element B[col][n], window folded in.
//  col 0..511   : bin=col+1 cos term, amp (bin==512 ? 1 : 2)/1024
//  col 512      : DC, 1/1024
//  col 513..515 : zero pad
//  col 516..1027: bin=col-515 -sin term, amp 2/1024 (Nyquist sin == 0)
//  col 1028..1031: zero pad
__device__ __forceinline__ float basis_val(int col, int n) {
    float w = (0.5f - 0.5f * cosf(TWO_PI * (float)n * (1.0f / 1024.0f))) * (1.0f / 1.5f);
    float v = 0.0f;
    if (col < 512) {
        int bin = col + 1;
        float amp = (bin == 512 ? 1.0f : 2.0f) * (1.0f / 1024.0f);
        float ang = TWO_PI * (float)((bin * n) & 1023) * (1.0f / 1024.0f); // exact int reduction
        v = amp * cosf(ang);
    } else if (col == 512) {
        v = 1.0f / 1024.0f;
    } else if (col >= 516 && col < 1028) {
        int bin = col - 515;
        float amp = (bin == 512 ? 1.0f : 2.0f) * (1.0f / 1024.0f);
        float ang = TWO_PI * (float)((bin * n) & 1023) * (1.0f / 1024.0f);
        v = -amp * sinf(ang);
    }
    return v * w;
}

// Build the basis directly in WMMA-B fragment order:
//   bswz[(nt*KCHUNKS + kc)*32 + lane] = { B[k0+d][n], B[k0+d+2][n] }
// with k0 = 4*kc, d = (lane<16 ? 0 : 1), n = nt*16 + (lane&15).
__global__ void build_basis_kernel(v2f* __restrict__ bswz) {
    int idx = blockIdx.x * blockDim.x + threadIdx.x;
    if (idx >= NTILES * KCHUNKS * 32) return;
    int lane = idx & 31;
    int kc   = (idx >> 5) % KCHUNKS;
    int nt   = idx / (KCHUNKS * 32);
    int n  = nt * 16 + (lane & 15);
    int k0 = kc * 4 + ((lane < 16) ? 0 : 1);
    v2f o;
    o.x = basis_val(k0,     n);
    o.y = basis_val(k0 + 2, n);
    bswz[idx] = o;
}

// Synchronous A-chunk staging with the bin remap (used for the two chunks
// that cross the DC/Nyquist/zero-pad seams, and as full fallback w/o TDM).
__device__ __forceinline__ void stage_sync(float (*AshB)[132], int tid, int mb, int chunk,
                                           const float* __restrict__ re,
                                           const float* __restrict__ im) {
    const int c0 = chunk * 128;
    const int cc = (chunk == 8) ? 8 : 128;
    for (int i = tid; i < 32 * cc; i += 256) {
        int row = (cc == 128) ? (i >> 7) : (i >> 3);
        int cl  = (cc == 128) ? (i & 127) : (i & 7);
        int col = c0 + cl;
        int rg  = mb * 32 + row;
        float v = 0.0f;
        if (col < 512)                      v = re[(size_t)rg * BINS + col + 1];
        else if (col == 512)                v = re[(size_t)rg * BINS];
        else if (col >= 516 && col < 1028)  v = im[(size_t)rg * BINS + (col - 515)];
        AshB[row][cl] = v;
    }
}

#if HAVE_TDM
// One TDM descriptor: 2D tile 128 (dim0) x 32 rows (dim1) of 4-byte elements,
// global row stride 513 elements, hardware LDS padding of 4 DWORDs after
// every 128 DWORDs stored -> LDS row stride 132 floats (conflict-free b64
// fragment reads). Issued by wave 0 only; tracked with TENSORcnt.
__device__ __forceinline__ void tdm_issue(float (*AshB)[132], int mb, int chunk,
                                          const float* __restrict__ re,
                                          const float* __restrict__ im) {
    const int c0 = chunk * 128;
    const float* src = (chunk < 4)
        ? (re + (size_t)mb * 32 * BINS + (size_t)(c0 + 1))     // real bins c0+1 ..
        : (im + (size_t)mb * 32 * BINS + (size_t)(c0 - 515));  // imag bins c0-515 ..
    const unsigned long long ga = (unsigned long long)(uintptr_t)src;
    // Generic LDS pointer low 32 bits == wave-relative LDS byte offset.
    const unsigned int lds = (unsigned int)(uintptr_t)(void*)AshB;

    // D# group 0: count=1 | lds_addr | global_addr[56:0] | type=2 at [127:126]
    u32x4 g0 = { 1u,
                 lds,
                 (unsigned int)ga,
                 (unsigned int)((ga >> 32) & 0x01FFFFFFu) | 0x80000000u };

    // D# group 1:
    //  dw0: data_size=2 (4B) | pad_enable | pad_interval=6 (128 DW) | pad_amount=3 (4 DW)
    //  tensor_dim0=128, tensor_dim1=32, tile_dim0=128, tile_dim1=32, tile_dim2=0
    //  tensor_dim0_stride=513, tensor_dim1_stride=0
    i32x8 g1 = { (int)((2u << 16) | (1u << 20) | (6u << 22) | (3u << 25)),
                 (int)(128u << 16),          // [63:48]  tensor_dim0 lo16
                 (int)(32u << 16),           // [95:80]  tensor_dim1 lo16 (dim0 hi16 = 0)
                 (int)(128u << 16),          // [127:112] tile_dim0   (dim1 hi16 = 0)
                 (int)(32u),                 // [143:128] tile_dim1, tile_dim2 = 0
                 (int)(513u),                // [191:160] tensor_dim0_stride lo32
                 0,                          // stride0 hi16 = 0, stride1 lo16 = 0
                 0 };

    i32x4 gz4 = { 0, 0, 0, 0 };              // groups 2/3 unused for 2D tensor
    i32x8 gz8 = { 0, 0, 0, 0, 0, 0, 0, 0 };  // extra descriptor words (6-arg form)
    __builtin_amdgcn_tensor_load_to_lds(g0, g1, gz4, gz4, gz8, 0);
}
#endif

// Stage chunk into the given LDS buffer: TDM for "pure" chunks, sync otherwise.
__device__ __forceinline__ void stage_chunk(float (*AshB)[132], int tid, int wave, int mb,
                                            int chunk,
                                            const float* __restrict__ re,
                                            const float* __restrict__ im) {
#if HAVE_TDM
    if (chunk != 4 && chunk != 8) {          // pure contiguous-bin tile
        if (wave == 0) tdm_issue(AshB, mb, chunk, re, im);
        return;
    }
#endif
    stage_sync(AshB, tid, mb, chunk, re, im);
}

// GEMM + overlap-add.
// Block: 256 threads (8 waves). Block owns 32 M-rows (frames), grid.y picks a
// group of 16 n-tiles; wave w handles n-tiles {2w, 2w+1} of that group.
// Each wave: 2 M-subtiles x 2 N-tiles of 16x16 fp32 accumulators.
__global__ void __launch_bounds__(256)
istft_wmma_kernel(const float* __restrict__ re, const float* __restrict__ im,
                  const v2f* __restrict__ bswz, float* __restrict__ out) {
    __shared__ float Ash[2][32][132];        // double-buffered A K-chunk

    const int tid  = threadIdx.x;
    const int lane = tid & 31;
    const int wave = tid >> 5;
    const int mb   = blockIdx.x;             // 0..1023 : rows mb*32 .. mb*32+31
    const int ng   = blockIdx.y;             // 0..3    : n-tile group
    const int half = (lane < 16) ? 0 : 1;
    const int mi   = lane & 15;

    v8f acc00 = {}, acc01 = {}, acc10 = {}, acc11 = {};

    const int nt0 = ng * 16 + wave * 2;
    const v2f* bp0 = bswz + (size_t)nt0 * KCHUNKS * 32 + lane;
    const v2f* bp1 = bp0 + (size_t)KCHUNKS * 32;

    // Prologue: stage chunk 0 into buffer 0.
    stage_chunk(Ash[0], tid, wave, mb, 0, re, im);

    for (int chunk = 0; chunk < 9; ++chunk) {
        const int cb = chunk & 1;
#if HAVE_TDM
        if (wave == 0) __builtin_amdgcn_s_wait_tensorcnt(0);  // chunk's TDM landed
#endif
        __syncthreads();                      // staged data visible to all waves
        if (chunk < 8)                        // prefetch next chunk into other buffer
            stage_chunk(Ash[cb ^ 1], tid, wave, mb, chunk + 1, re, im);

        const int c0  = chunk * 128;
        const int kcn = (chunk == 8) ? 2 : 32;
        for (int kk = 0; kk < kcn; ++kk) {
            const int kc = (c0 >> 2) + kk;
            const int kl = kk * 4 + half * 2;  // A frag: K rows {kl, kl+1}
            v2f a0 = *(const v2f*)&Ash[cb][mi][kl];
            v2f a1 = *(const v2f*)&Ash[cb][16 + mi][kl];
            v2f b0 = bp0[kc * 32];
            v2f b1 = bp1[kc * 32];
            __builtin_prefetch(bp0 + (kc + 8) * 32, 0, 3);  // WGP-scope B stream-ahead
            acc00 = __builtin_amdgcn_wmma_f32_16x16x4_f32(false, a0, false, b0, (short)0, acc00, false, false);
            acc01 = __builtin_amdgcn_wmma_f32_16x16x4_f32(false, a0, false, b1, (short)0, acc01, false, false);
            acc10 = __builtin_amdgcn_wmma_f32_16x16x4_f32(false, a1, false, b0, (short)0, acc10, false, false);
            acc11 = __builtin_amdgcn_wmma_f32_16x16x4_f32(false, a1, false, b1, (short)0, acc11, false, false);
        }
    }

    // Overlap-add with trim: out[b, f*256 + n - 512] += frame value.
    // C/D layout: VGPR r -> M = r + 8*half, N = lane&15.
    const int nA = nt0 * 16 + mi;
    const int nB = nA + 16;
    for (int sub = 0; sub < 2; ++sub) {
        v8f accA = sub ? acc10 : acc00;
        v8f accB = sub ? acc11 : acc01;
        int rg_base = mb * 32 + sub * 16;
        for (int r = 0; r < 8; ++r) {
            int rg = rg_base + r + half * 8;
            int b  = rg >> 9;          // / FRAMES
            int f  = rg & 511;         // % FRAMES
            int pb = f * STRIDE_ - 512;
            int tA = pb + nA;
            int tB = pb + nB;
            if (tA >= 0 && tA < OUT_T) atomicAdd(&out[(size_t)b * OUT_T + tA], accA[r]);
            if (tB >= 0 && tB < OUT_T) atomicAdd(&out[(size_t)b * OUT_T + tB], accB[r]);
        }
    }
}

extern "C" void kernel_launch(void* const* d_in, const int* in_sizes, int n_in,
                              void* d_out, int out_size, void* d_ws, size_t ws_size,
                              hipStream_t stream) {
    const float* re = (const float*)d_in[0];
    const float* im = (const float*)d_in[1];
    float* out = (float*)d_out;
    v2f* basis = (v2f*)d_ws;   // needs 1032*1024*4 = 4.03 MiB of scratch

    // Zero the overlap-add target (graph-capturable).
    (void)hipMemsetAsync(d_out, 0, (size_t)out_size * sizeof(float), stream);

    // 1) Build the windowed DFT basis, pre-swizzled into WMMA B-fragment order.
    int total = NTILES * KCHUNKS * 32;
    build_basis_kernel<<<(total + 255) / 256, 256, 0, stream>>>(basis);

    // 2) GEMM (fp32 WMMA) + windowed overlap-add.
    dim3 grid(1024, 4);
    istft_wmma_kernel<<<grid, 256, 0, stream>>>(re, im, (const v2f*)basis, out);
}